// DotAttention_72378788872872
// MI455X (gfx1250) — compile-verified
//
#include <hip/hip_runtime.h>
#include <math.h>

typedef __attribute__((ext_vector_type(2))) float v2f;
typedef __attribute__((ext_vector_type(8))) float v8f;

// types matching the async-to-LDS builtin signature (from the compiler diagnostic)
typedef int v4i_vs __attribute__((__vector_size__(16)));
typedef __attribute__((address_space(1))) v4i_vs gv4i;
typedef __attribute__((address_space(3))) v4i_vs lv4i;

#define S_LEN 2048
#define D_DIM 128
#define BR    128            // q rows per workgroup (8 waves x 16)
#define BC    16             // kv rows per LDS tile
#define NITER (S_LEN / BC)
#define KST   132            // K tile row stride (dwords): 132%64=4 -> conflict-free b64 reads
#define VST   136            // V tile row stride (dwords): 2*136%64=16 -> halves on disjoint banks
#define PST   20             // P scratch row stride (dwords)

__device__ __forceinline__ v8f wmma4(v2f a, v2f b, v8f c) {
  return __builtin_amdgcn_wmma_f32_16x16x4_f32(
      false, a, false, b, (short)0, c, false, false);
}

// 16B global -> LDS async copy (per lane), tracked by ASYNCcnt
__device__ __forceinline__ void async_copy_b128(const float* g, float* l) {
#if __has_builtin(__builtin_amdgcn_global_load_async_to_lds_b128)
  __builtin_amdgcn_global_load_async_to_lds_b128(
      (gv4i*)(unsigned long long)(uintptr_t)g,
      (lv4i*)(unsigned)(uintptr_t)l,
      0, 0);
#else
  unsigned loff = (unsigned)(uintptr_t)l;   // low 32 bits of generic LDS ptr = LDS offset
  asm volatile("global_load_async_to_lds_b128 %0, %1, off"
               :: "v"(loff), "v"(g) : "memory");
#endif
}

__device__ __forceinline__ void wait_async0() {
#if __has_builtin(__builtin_amdgcn_s_wait_asynccnt)
  __builtin_amdgcn_s_wait_asynccnt(0);
#else
  asm volatile("s_wait_asynccnt 0x0" ::: "memory");
#endif
}

__device__ __forceinline__ void issue_tile(const float* kg, const float* vg,
                                           float* kb, float* vb, int tid) {
  // tile = 16x128 floats = 8KB per tensor; 256 threads x 2 chunks of 16B each
#pragma unroll
  for (int i = 0; i < 2; ++i) {
    int flat = (tid + i * 256) * 4;
    int row  = flat >> 7;
    int col  = flat & 127;
    async_copy_b128(kg + flat, kb + row * KST + col);
    async_copy_b128(vg + flat, vb + row * VST + col);
  }
}

__global__ __launch_bounds__(256)
void DotAttention_fa_kernel(const float* __restrict__ Q,
                            const float* __restrict__ K,
                            const float* __restrict__ V,
                            float* __restrict__ O) {
  __shared__ __align__(16) float kbuf[2][BC * KST];
  __shared__ __align__(16) float vbuf[2][BC * VST];
  __shared__ float pbuf[8 * 16 * PST];

  const int tid  = threadIdx.x;
  const int wave = tid >> 5;
  const int lane = tid & 31;
  const int n16  = lane & 15;
  const int half = lane >> 4;
  const int kh   = half * 2;

  const long bh    = blockIdx.y;
  const long base  = bh * (long)(S_LEN * D_DIM);
  const int  qrow0 = blockIdx.x * BR + wave * 16;

  // ---- resident Q slice as WMMA A-fragments, 1/128 scale folded in ----
  v2f aq[32];
  {
    const float* qp = Q + base + (long)(qrow0 + n16) * D_DIM + kh;
#pragma unroll
    for (int kk = 0; kk < 32; ++kk) {
      v2f t = *(const v2f*)(qp + kk * 4);
      aq[kk].x = t.x * (1.0f / 128.0f);
      aq[kk].y = t.y * (1.0f / 128.0f);
    }
  }

  v8f o[8];
#pragma unroll
  for (int t = 0; t < 8; ++t) o[t] = (v8f)0.0f;
  float mrow[8], lrow[8];
#pragma unroll
  for (int r = 0; r < 8; ++r) { mrow[r] = -INFINITY; lrow[r] = 0.0f; }

  float* pw = pbuf + wave * 16 * PST;

  // ---- software pipeline: prefetch tile 0 ----
  issue_tile(K + base, V + base, kbuf[0], vbuf[0], tid);

  for (int j = 0; j < NITER; ++j) {
    const int cur = j & 1;

    wait_async0();        // own async writes for tile j complete
    __syncthreads();      // publish tile j to all waves; frees buf[cur^1]

    if (j + 1 < NITER) {  // prefetch tile j+1 while computing tile j
      const long off = base + (long)(j + 1) * BC * D_DIM;
      issue_tile(K + off, V + off, kbuf[cur ^ 1], vbuf[cur ^ 1], tid);
    }

    // ---- S = (Q*scale) @ K_tile^T (16x16) ----
    v8f s0 = (v8f)0.0f, s1 = (v8f)0.0f;
    const float* kbp = kbuf[cur] + n16 * KST + kh;
#pragma unroll
    for (int kk = 0; kk < 32; kk += 2) {
      v2f b0 = *(const v2f*)(kbp + kk * 4);
      s0 = wmma4(aq[kk], b0, s0);
      v2f b1 = *(const v2f*)(kbp + kk * 4 + 4);
      s1 = wmma4(aq[kk + 1], b1, s1);
    }

    // ---- online softmax (rows r / r+8 in the two lane halves) ----
    float alpha[8];
#pragma unroll
    for (int r = 0; r < 8; ++r) {
      float sv = s0[r] + s1[r];
      float rm = sv;
      rm = fmaxf(rm, __shfl_xor(rm, 1));
      rm = fmaxf(rm, __shfl_xor(rm, 2));
      rm = fmaxf(rm, __shfl_xor(rm, 4));
      rm = fmaxf(rm, __shfl_xor(rm, 8));
      float mnew = fmaxf(mrow[r], rm);
      float a    = __expf(mrow[r] - mnew);
      float p    = __expf(sv - mnew);
      float rs = p;
      rs += __shfl_xor(rs, 1);
      rs += __shfl_xor(rs, 2);
      rs += __shfl_xor(rs, 4);
      rs += __shfl_xor(rs, 8);
      lrow[r]  = lrow[r] * a + rs;
      mrow[r]  = mnew;
      alpha[r] = a;
      pw[(r + 8 * half) * PST + n16] = p;   // stash P (C-layout)
    }
#pragma unroll
    for (int t = 0; t < 8; ++t) {
#pragma unroll
      for (int r = 0; r < 8; ++r) o[t][r] *= alpha[r];
    }

    // ---- reload P as A-fragments (same-wave DS ops are in-order) ----
    v2f pa[4];
#pragma unroll
    for (int k2 = 0; k2 < 4; ++k2)
      pa[k2] = *(const v2f*)(pw + n16 * PST + k2 * 4 + kh);

    // ---- O += P @ V_tile (V row-major; b.x/b.y one row apart) ----
#pragma unroll
    for (int t = 0; t < 8; ++t) {
      const float* vbp = vbuf[cur] + kh * VST + t * 16 + n16;
#pragma unroll
      for (int k2 = 0; k2 < 4; ++k2) {
        v2f b;
        b.x = vbp[(k2 * 4) * VST];
        b.y = vbp[(k2 * 4 + 1) * VST];
        o[t] = wmma4(pa[k2], b, o[t]);
      }
    }
  }

  // ---- epilogue: normalize and store ----
  float rinv[8];
#pragma unroll
  for (int r = 0; r < 8; ++r) rinv[r] = 1.0f / lrow[r];
#pragma unroll
  for (int t = 0; t < 8; ++t) {
#pragma unroll
    for (int r = 0; r < 8; ++r) {
      int m = r + 8 * half;
      O[base + (long)(qrow0 + m) * D_DIM + t * 16 + n16] = o[t][r] * rinv[r];
    }
  }
}

extern "C" void kernel_launch(void* const* d_in, const int* in_sizes, int n_in,
                              void* d_out, int out_size, void* d_ws, size_t ws_size,
                              hipStream_t stream) {
  const float* q = (const float*)d_in[0];
  const float* k = (const float*)d_in[1];
  const float* v = (const float*)d_in[2];
  float* out = (float*)d_out;
  int bh = in_sizes[0] / (S_LEN * D_DIM);   // B*H = 64
  dim3 grid(S_LEN / BR, bh);
  DotAttention_fa_kernel<<<grid, 256, 0, stream>>>(q, k, v, out);
}